// JointsOHKMMSELoss_10196252360784
// MI455X (gfx1250) — compile-verified
//
#include <hip/hip_runtime.h>

// JointsOHKMMSELoss for MI455X (gfx1250, wave32).
// Stage 1: memory-bound streaming reduction (160 MB @ 23.3 TB/s ~ 7us floor).
//   - 128-bit (ext_vector float4) non-temporal loads, 256 thr = 8 waves/block.
//   - v_wmma_f32_16x16x4_f32 with B == ones used as a matrix-pipe reduction:
//     D(m,n) = sum_k A(m,k) + C(m,n)  -> sum of all C elements = 16 * total.
// Stage 2: tiny top-8-of-17 selection + mean, single block.

typedef float v2f __attribute__((ext_vector_type(2)));
typedef float v4f __attribute__((ext_vector_type(4)));
typedef float v8f __attribute__((ext_vector_type(8)));

#define K_JOINTS 17
#define TOPK_N   8
#define T1       256   // stage-1 block size (8 wave32s)

__global__ __launch_bounds__(T1) void per_joint_mse_kernel(
    const float* __restrict__ output,
    const float* __restrict__ target,
    const float* __restrict__ tw,
    float* __restrict__ per_joint,
    int hw)
{
  const int bk  = blockIdx.x;
  const int tid = threadIdx.x;
  const long long base = (long long)bk * (long long)hw;
  const v4f* __restrict__ o4 = (const v4f*)(output + base);
  const v4f* __restrict__ t4 = (const v4f*)(target + base);
  const int hw4   = hw >> 2;        // hw is a multiple of 4 (96*96 = 9216)
  const int iters = hw4 / T1;       // uniform trip count -> EXEC all-1s in WMMA loop
  const int rem_i = iters * T1 + tid;

  v8f c = {0.f, 0.f, 0.f, 0.f, 0.f, 0.f, 0.f, 0.f};
  v2f bones; bones[0] = 1.0f; bones[1] = 1.0f;   // B = ones -> WMMA does row sums

  for (int it = 0; it < iters; ++it) {
    const int i = tid + it * T1;
    v4f o = __builtin_nontemporal_load(&o4[i]);
    v4f t = __builtin_nontemporal_load(&t4[i]);
    const float d0 = o[0] - t[0];
    const float d1 = o[1] - t[1];
    const float d2 = o[2] - t[2];
    const float d3 = o[3] - t[3];
    v2f a;
    a[0] = d0 * d0 + d1 * d1;
    a[1] = d2 * d2 + d3 * d3;
    // 8-arg pattern: (neg_a, A, neg_b, B, c_mod, C, reuse_a, reuse_b)
    c = __builtin_amdgcn_wmma_f32_16x16x4_f32(
        false, a, false, bones, (short)0, c, false, false);
  }

  // Remainder (not taken for hw4 % 256 == 0) handled with plain VALU so the
  // WMMA loop above never runs with partial EXEC.
  float extra = 0.f;
  if (rem_i < hw4) {
    v4f o = __builtin_nontemporal_load(&o4[rem_i]);
    v4f t = __builtin_nontemporal_load(&t4[rem_i]);
    const float d0 = o[0] - t[0];
    const float d1 = o[1] - t[1];
    const float d2 = o[2] - t[2];
    const float d3 = o[3] - t[3];
    extra = d0 * d0 + d1 * d1 + d2 * d2 + d3 * d3;
  }

  // Each row-sum is replicated across 16 columns of C -> divide by 16.
  float lane = (c[0] + c[1] + c[2] + c[3] + c[4] + c[5] + c[6] + c[7]) * (1.0f / 16.0f)
             + extra;

  // wave32 butterfly reduction
  for (int off = 16; off > 0; off >>= 1)
    lane += __shfl_down(lane, off, 32);

  __shared__ float wsum[T1 / 32];
  if ((tid & 31) == 0) wsum[tid >> 5] = lane;
  __syncthreads();

  if (tid == 0) {
    float s = 0.f;
    #pragma unroll
    for (int w = 0; w < T1 / 32; ++w) s += wsum[w];
    const float wgt = tw[bk];
    // diff = (pred - gt) * w  ->  mean(diff^2) = w^2 * sum(d^2) / HW
    per_joint[bk] = s * wgt * wgt / (float)hw;
  }
}

__global__ __launch_bounds__(128) void ohkm_topk_mean_kernel(
    const float* __restrict__ per_joint,
    float* __restrict__ out,
    int B)
{
  const int tid = threadIdx.x;
  float acc = 0.f;

  for (int b = tid; b < B; b += 128) {
    float v[K_JOINTS];
    const float* row = per_joint + b * K_JOINTS;
    #pragma unroll
    for (int k = 0; k < K_JOINTS; ++k) v[k] = row[k];

    float tsum = 0.f;
    #pragma unroll
    for (int t = 0; t < TOPK_N; ++t) {
      int   mi = 0;
      float mv = v[0];
      #pragma unroll
      for (int k = 1; k < K_JOINTS; ++k) {
        if (v[k] > mv) { mv = v[k]; mi = k; }
      }
      tsum += mv;
      v[mi] = -3.402823466e38f;
    }
    acc += tsum;
  }

  for (int off = 16; off > 0; off >>= 1)
    acc += __shfl_down(acc, off, 32);

  __shared__ float wsum[4];
  if ((tid & 31) == 0) wsum[tid >> 5] = acc;
  __syncthreads();

  if (tid == 0) {
    float s = wsum[0] + wsum[1] + wsum[2] + wsum[3];
    out[0] = s / (float)(B * TOPK_N);   // LOSS_WEIGHT == 1.0
  }
}

extern "C" void kernel_launch(void* const* d_in, const int* in_sizes, int n_in,
                              void* d_out, int out_size, void* d_ws, size_t ws_size,
                              hipStream_t stream) {
  const float* output = (const float*)d_in[0];   // [B, K, H, W] f32
  const float* target = (const float*)d_in[1];   // [B, K, H, W] f32
  const float* tw     = (const float*)d_in[2];   // [B, K, 1]    f32
  float* out = (float*)d_out;                    // scalar f32

  const int total = in_sizes[0];
  const int BK    = in_sizes[2];                 // B * K
  const int hw    = total / BK;                  // 9216
  const int B     = BK / K_JOINTS;               // 128

  float* per_joint = (float*)d_ws;               // BK floats of scratch

  per_joint_mse_kernel<<<BK, T1, 0, stream>>>(output, target, tw, per_joint, hw);
  ohkm_topk_mean_kernel<<<1, 128, 0, stream>>>(per_joint, out, B);
}